// TTrain_6519760355424
// MI455X (gfx1250) — compile-verified
//
#include <hip/hip_runtime.h>

// Tensor-train (MPS) contraction for MI455X / gfx1250 (wave32, WMMA).
//
//  probs: per-batch chain v <- v @ core[x_t]  (batched GEMV chain, VALU fp32,
//         shuffle-broadcast state, core is L2-resident)
//  norm : c' = sum_i G_i^T (c G_i) transfer chain, fp32 WMMA 16x16x4,
//         single 32-wave workgroup, ~290KB of the 320KB LDS holds c, c', and
//         the stacked intermediate U = vstack(c @ G_i).

#define D_CAT  16
#define BOND   64
#define BATCH  1024
#define SEQLEN 512
#define PD     68   // padded row stride (floats) for c buffers: kills LDS bank conflicts

typedef __attribute__((ext_vector_type(2))) float v2f;
typedef __attribute__((ext_vector_type(8))) float v8f;

__device__ __forceinline__ v8f wmma4(v2f a, v2f b, v8f c) {
  // V_WMMA_F32_16X16X4_F32 : D = A(16x4) * B(4x16) + C(16x16), all fp32
  return __builtin_amdgcn_wmma_f32_16x16x4_f32(
      /*neg_a=*/false, a, /*neg_b=*/false, b,
      /*c_mod=*/(short)0, c, /*reuse_a=*/false, /*reuse_b=*/false);
}

// ---------------------------------------------------------------------------
// norm path: one workgroup, 1024 threads = 32 wave32s on one WGP.
//   c0 = A^T A,  A[i,k] = sum_j lb[j] G_i[j,k]
//   repeat 511x:  U_i = c @ G_i   (stage 1),   c' = sum_i G_i^T U_i (stage 2)
//   norm = rb^T c rb
// ---------------------------------------------------------------------------
__device__ __forceinline__ void norm_path(const float* __restrict__ core,
                                          const float* __restrict__ lb,
                                          const float* __restrict__ rb,
                                          float* __restrict__ out_norm,
                                          float* sU,   // [D_CAT * BOND * BOND]
                                          float* sc,   // [BOND * PD]
                                          float* scn)  // [BOND * PD]
{
  const int tid  = threadIdx.x;      // 0..1023
  const int wave = tid >> 5;         // 0..31
  const int lane = tid & 31;
  const int lo   = lane & 15;        // lane%16
  const int hi   = lane >> 4;        // 0 or 1

  // ---- A[i][k] = sum_j lb[j] * G_i[j][k]  (scratch in scn) ----
  {
    const int i = tid >> 6, k = tid & 63;
    const float* Gi = core + i * (BOND * BOND);
    float s = 0.f;
#pragma unroll 4
    for (int j = 0; j < BOND; ++j) s = fmaf(lb[j], Gi[j * BOND + k], s);
    scn[i * BOND + k] = s;
  }
  __syncthreads();
  // ---- c0[j][k] = sum_i A[i][j] A[i][k] ----
  for (int e = tid; e < BOND * BOND; e += 1024) {
    const int j = e >> 6, k = e & 63;
    float s = 0.f;
#pragma unroll
    for (int i = 0; i < D_CAT; ++i) s = fmaf(scn[i * BOND + j], scn[i * BOND + k], s);
    sc[j * PD + k] = s;
  }
  __syncthreads();

  // ---- 511 transfer-matrix steps ----
  for (int step = 0; step < SEQLEN - 1; ++step) {
    // zero next-c accumulator (stage 2 uses ds_add_f32 atomics)
    for (int e = tid; e < BOND * BOND; e += 1024)
      scn[(e >> 6) * PD + (e & 63)] = 0.f;

    // ---- stage 1: U_i = c @ G_i ; symbol i = wave/2, 8 tiles of 16x16 each ----
    {
      const int i = wave >> 1;
      const float* Gi = core + i * (BOND * BOND);
      float* Ui = sU + i * (BOND * BOND);
      for (int t8 = 0; t8 < 8; ++t8) {
        const int t  = ((wave & 1) << 3) + t8;   // tile 0..15
        const int mt = t >> 2, nt = t & 3;
        v8f acc = {};
#pragma unroll 4
        for (int kb = 0; kb < 16; ++kb) {
          const int k0 = kb * 4 + 2 * hi;
          v2f a, b;
          a.x = sc[(mt * 16 + lo) * PD + k0];      // A = c tile (16x4)
          a.y = sc[(mt * 16 + lo) * PD + k0 + 1];
          b.x = Gi[(k0)     * BOND + nt * 16 + lo]; // B = G tile (4x16)
          b.y = Gi[(k0 + 1) * BOND + nt * 16 + lo];
          acc = wmma4(a, b, acc);
        }
#pragma unroll
        for (int r = 0; r < 8; ++r)
          Ui[(mt * 16 + r + 8 * hi) * BOND + nt * 16 + lo] = acc[r];
      }
    }
    __syncthreads();

    // ---- stage 2: c' = sum_i G_i^T @ U_i  (K = 16*64); 2 waves per tile ----
    {
      const int t  = wave >> 1;                 // output tile 0..15
      const int mt = t >> 2, nt = t & 3;
      const int i0 = (wave & 1) << 3;           // this wave sums 8 symbols
      v8f acc = {};
      for (int ii = 0; ii < 8; ++ii) {
        const float* Gi = core + (i0 + ii) * (BOND * BOND);
        const float* Ui = sU   + (i0 + ii) * (BOND * BOND);
#pragma unroll 4
        for (int kb = 0; kb < 16; ++kb) {
          const int k0 = kb * 4 + 2 * hi;
          v2f a, b;
          a.x = Gi[(k0)     * BOND + mt * 16 + lo]; // A = G^T tile: A[m,k]=G[k,m]
          a.y = Gi[(k0 + 1) * BOND + mt * 16 + lo];
          b.x = Ui[(k0)     * BOND + nt * 16 + lo]; // B = U tile (4x16)
          b.y = Ui[(k0 + 1) * BOND + nt * 16 + lo];
          acc = wmma4(a, b, acc);
        }
      }
#pragma unroll
      for (int r = 0; r < 8; ++r)
        atomicAdd(&scn[(mt * 16 + r + 8 * hi) * PD + nt * 16 + lo], acc[r]);
    }
    __syncthreads();

    // commit c <- c'
    for (int e = tid; e < BOND * BOND; e += 1024) {
      const int r = e >> 6, cc = e & 63;
      sc[r * PD + cc] = scn[r * PD + cc];
    }
    __syncthreads();
  }

  // ---- norm = rb^T c rb ----
  if (tid < BOND) {
    float s = 0.f;
#pragma unroll 4
    for (int k = 0; k < BOND; ++k) s = fmaf(sc[tid * PD + k], rb[k], s);
    scn[tid] = s * rb[tid];
  }
  __syncthreads();
  if (tid == 0) {
    float s = 0.f;
    for (int k = 0; k < BOND; ++k) s += scn[k];
    out_norm[0] = s;
  }
}

// ---------------------------------------------------------------------------
// probs path: one wave32 per batch element; state v(64) held 2 floats/lane,
// broadcast via shuffles; core rows read as float2 (L2-resident, 256KB).
// ---------------------------------------------------------------------------
__device__ __forceinline__ void probs_path(const int* __restrict__ x,
                                           const float* __restrict__ core,
                                           const float* __restrict__ lb,
                                           const float* __restrict__ rb,
                                           float* __restrict__ out, int blk)
{
  const int wave = threadIdx.x >> 5;
  const int lane = threadIdx.x & 31;
  const int b    = blk * 32 + wave;
  const int j0   = lane * 2;                  // this lane owns columns j0, j0+1

  float vx0 = lb[j0], vx1 = lb[j0 + 1];
  const int* xb = x + b * SEQLEN;

  for (int t = 0; t < SEQLEN; ++t) {
    const float* M = core + xb[t] * (BOND * BOND);
    float nx = 0.f, ny = 0.f;
#pragma unroll 8
    for (int kk = 0; kk < 32; ++kk) {
      const float va = __shfl(vx0, kk);       // v[2kk]
      const float vb = __shfl(vx1, kk);       // v[2kk+1]
      const float2 g0 = *reinterpret_cast<const float2*>(M + (2 * kk)     * BOND + j0);
      const float2 g1 = *reinterpret_cast<const float2*>(M + (2 * kk + 1) * BOND + j0);
      nx = fmaf(va, g0.x, nx); ny = fmaf(va, g0.y, ny);
      nx = fmaf(vb, g1.x, nx); ny = fmaf(vb, g1.y, ny);
    }
    vx0 = nx; vx1 = ny;
  }

  float p = vx0 * rb[j0] + vx1 * rb[j0 + 1];
#pragma unroll
  for (int off = 16; off > 0; off >>= 1) p += __shfl_xor(p, off);
  if (lane == 0) out[b] = p;
}

// ---------------------------------------------------------------------------
__global__ __launch_bounds__(1024, 1)
void ttrain_kernel(const int* __restrict__ x, const float* __restrict__ core,
                   const float* __restrict__ lb, const float* __restrict__ rb,
                   float* __restrict__ out)
{
  // ~290KB of the 320KB CDNA5 LDS (norm block only; probs blocks ignore it)
  __shared__ float sU[D_CAT * BOND * BOND];   // 256 KB: stacked U_i
  __shared__ float sc[BOND * PD];             // 17 KB: current c (padded)
  __shared__ float scn[BOND * PD];            // 17 KB: next c (padded)

  if (blockIdx.x == 0) {
    norm_path(core, lb, rb, out + BATCH, sU, sc, scn);
  } else {
    probs_path(x, core, lb, rb, out, (int)blockIdx.x - 1);
  }
}

extern "C" void kernel_launch(void* const* d_in, const int* in_sizes, int n_in,
                              void* d_out, int out_size, void* d_ws, size_t ws_size,
                              hipStream_t stream) {
  (void)in_sizes; (void)n_in; (void)out_size; (void)d_ws; (void)ws_size;
  const int*   x    = (const int*)d_in[0];
  const float* core = (const float*)d_in[1];
  const float* lb   = (const float*)d_in[2];
  const float* rb   = (const float*)d_in[3];
  float* out = (float*)d_out;   // [0..1023] probs, [1024] norm

  dim3 grid(1 + BATCH / 32);    // block 0: norm chain; blocks 1..32: probs
  ttrain_kernel<<<grid, 1024, 0, stream>>>(x, core, lb, rb, out);
}